// TopK_36653250904599
// MI455X (gfx1250) — compile-verified
//
#include <hip/hip_runtime.h>
#include <hip/hip_bf16.h>
#include <stdint.h>

#define DIM_B   16
#define DIM_C   256
#define DIM_H   128
#define DIM_W   128
#define HW      (DIM_H * DIM_W)     // 16384 elements per row
#define ROWS    (DIM_B * DIM_C)     // 4096 rows
#define TOPK    16
#define NT      256                 // 8 wave32 per block
#define NWAVE   (NT / 32)
#define HALF    (HW / 2)            // 8192 elements per TDM chunk (32 KB)
#define NEG_HUGE (-3.402823466e38f)

typedef unsigned int u32x4 __attribute__((ext_vector_type(4)));
typedef int          i32x8 __attribute__((ext_vector_type(8)));
typedef int          i32x4 __attribute__((ext_vector_type(4)));

// LDS layout (reuses the 64KB row buffer after phase 1):
//   [0      .. 4351]  lv: per-thread sorted top-16 values, stride 17 (bank-conflict pad)
//   [4352   .. 8703]  li: per-thread sorted top-16 indices, stride 17
//   [8704   .. 8711]  per-wave sum-of-squares partials
//   [8712   .. 8719]  per-wave argmax value
//   [8720   .. 8727]  per-wave argmax global index
//   [8728   .. 8735]  per-wave argmax owner thread
//   [8736]            broadcast: winning thread of this round
#define LV_OFF   0
#define LI_OFF   4352
#define RED_OFF  8704
#define WMV_OFF  8712
#define WMI_OFF  8720
#define WMT_OFF  8728
#define BC_OFF   8736

// Issue one TDM descriptor: linear 1D tile of `nelem` f32 from gptr -> LDS ldsoff.
__device__ __forceinline__ void tdm_load_chunk(const float* gptr, uint32_t ldsoff,
                                               int nelem) {
  const uint64_t gaddr = (uint64_t)(uintptr_t)gptr;
  u32x4 g0;
  g0[0] = 1u;                                          // count=1 (valid), user mode
  g0[1] = ldsoff;                                      // lds_addr (bytes)
  g0[2] = (uint32_t)(gaddr & 0xFFFFFFFFull);           // global_addr[31:0]
  g0[3] = (uint32_t)((gaddr >> 32) & 0x1FFFFFFull)     // global_addr[56:32]
        | (2u << 30);                                  // type = 2 ("image")
  i32x8 g1;
  g1[0] = (int)(2u << 16);                // workgroup_mask=0, data_size=2 (4 bytes)
  g1[1] = (int)((uint32_t)nelem << 16);   // tensor_dim0[15:0]
  g1[2] = (int)(1u << 16);                // tensor_dim0[31:16]=0, tensor_dim1=1
  g1[3] = (int)((uint32_t)nelem << 16);   // tile_dim0 (16-bit field)
  g1[4] = 1;                              // tile_dim1=1, tile_dim2=0
  g1[5] = nelem;                          // tensor_dim0_stride
  g1[6] = 0;
  g1[7] = 0;
  i32x4 gz4 = {0, 0, 0, 0};               // groups 2/3 NULL => 2-group (<=2D) form
  i32x8 gz8 = {0, 0, 0, 0, 0, 0, 0, 0};
  __builtin_amdgcn_tensor_load_to_lds(g0, g1, gz4, gz4, gz8, /*cpol=*/0);
}

__global__ __launch_bounds__(NT)
void topk_l2_rows_kernel(const float* __restrict__ x,
                         float* __restrict__ out_val,
                         int*   __restrict__ out_coor) {
  __shared__ __align__(16) float rowbuf[HW];   // 64 KB

  const int row  = blockIdx.x;
  const int t    = threadIdx.x;
  const int lane = t & 31;
  const int wave = t >> 5;

  // ---- Phase 0: issue BOTH 32KB TDM chunks back-to-back (in-order per wave) ----
  if (t == 0) {
    const float* rowp = x + (size_t)row * HW;
    const uint32_t base = (uint32_t)(uintptr_t)rowbuf;   // low 32 bits = LDS offset
    tdm_load_chunk(rowp,        base,                 HALF);   // chunk 0
    tdm_load_chunk(rowp + HALF, base + HALF * 4u,     HALF);   // chunk 1
  }

  // ---- Phase 1: per-thread top-16 + sumsq, overlapped with chunk-1 DMA --------
  float tv[TOPK];
  int   ti[TOPK];
#pragma unroll
  for (int s = 0; s < TOPK; ++s) { tv[s] = NEG_HUGE; ti[s] = 0; }
  float sumsq = 0.0f;
  const float4* rb4 = (const float4*)rowbuf;

  auto insert = [&](float v, int idx) {
    if (v > tv[TOPK - 1]) {                  // fast reject: common case is 1 compare
      float cv = v; int ci = idx;
#pragma unroll
      for (int s = 0; s < TOPK; ++s) {
        if (cv > tv[s]) {
          float a = tv[s]; int b = ti[s];
          tv[s] = cv; ti[s] = ci;
          cv = a; ci = b;
        }
      }
    }
  };

  auto process_chunk = [&](int c) {
    for (int i = 0; i < HALF / (4 * NT); ++i) {          // 8 iters of ds_load_b128
      const int e4 = c * (HALF / 4) + i * NT + t;
      const float4 v = rb4[e4];
      sumsq = fmaf(v.x, v.x, sumsq);
      sumsq = fmaf(v.y, v.y, sumsq);
      sumsq = fmaf(v.z, v.z, sumsq);
      sumsq = fmaf(v.w, v.w, sumsq);
      const int base = e4 * 4;
      insert(v.x, base + 0);
      insert(v.y, base + 1);
      insert(v.z, base + 2);
      insert(v.w, base + 3);
    }
  };

  // TDM ops from one wave retire in order: tensorcnt<=1 means chunk 0 landed.
  __builtin_amdgcn_s_wait_tensorcnt(1);      // wave 0 waits; others pass (cnt==0)
  __syncthreads();
  process_chunk(0);                          // compute on chunk 0 while chunk 1 streams

  __builtin_amdgcn_s_wait_tensorcnt(0);      // chunk 1 landed
  __syncthreads();
  process_chunk(1);

  __syncthreads();                           // done reading rowbuf; safe to reuse LDS

  // ---- Phase 2: spill sorted lists + reduce sumsq -----------------------------
  float* lv  = rowbuf + LV_OFF;
  int*   li  = (int*)(rowbuf + LI_OFF);
  float* red = rowbuf + RED_OFF;
  float* wmv = rowbuf + WMV_OFF;
  int*   wmi = (int*)(rowbuf + WMI_OFF);
  int*   wmt = (int*)(rowbuf + WMT_OFF);
  int*   bc  = (int*)(rowbuf + BC_OFF);

#pragma unroll
  for (int s = 0; s < TOPK; ++s) {
    lv[t * 17 + s] = tv[s];
    li[t * 17 + s] = ti[s];
  }

#pragma unroll
  for (int off = 16; off >= 1; off >>= 1)    // wave32 reduction
    sumsq += __shfl_xor(sumsq, off, 32);
  if (lane == 0) red[wave] = sumsq;

  __syncthreads();

  float invn = 0.0f;
  if (t == 0) {
    float tot = 0.0f;
#pragma unroll
    for (int w = 0; w < NWAVE; ++w) tot += red[w];
    invn = rsqrtf(fmaxf(tot, 1e-12f));       // tf.nn.l2_normalize epsilon
  }

  // ---- Phase 3: 16 rounds of block-wide argmax merge --------------------------
  int p = 0;                                 // pointer into my sorted list
  for (int k = 0; k < TOPK; ++k) {
    float hval = (p < TOPK) ? lv[t * 17 + p] : NEG_HUGE;
    int   hidx = (p < TOPK) ? li[t * 17 + p] : 0;
    int   hsrc = t;
#pragma unroll
    for (int off = 16; off >= 1; off >>= 1) {
      float ov = __shfl_xor(hval, off, 32);
      int   oi = __shfl_xor(hidx, off, 32);
      int   os = __shfl_xor(hsrc, off, 32);
      if (ov > hval) { hval = ov; hidx = oi; hsrc = os; }
    }
    if (lane == 0) { wmv[wave] = hval; wmi[wave] = hidx; wmt[wave] = hsrc; }
    __syncthreads();

    if (t == 0) {
      float best = wmv[0]; int bi = wmi[0]; int bt = wmt[0];
#pragma unroll
      for (int w = 1; w < NWAVE; ++w)
        if (wmv[w] > best) { best = wmv[w]; bi = wmi[w]; bt = wmt[w]; }
      bc[0] = bt;
      const int o = row * TOPK + k;
      out_val[o]          = best * invn;        // normalized top-k value
      out_coor[o * 2 + 0] = bi >> 7;            // h = idx / W   (W = 128)
      out_coor[o * 2 + 1] = bi & (DIM_W - 1);   // w = idx % W
    }
    __syncthreads();

    if (t == bc[0]) p++;                     // winner advances its sorted list
    // next round's first __syncthreads orders the bc[0] rewrite after these reads
  }
}

extern "C" void kernel_launch(void* const* d_in, const int* in_sizes, int n_in,
                              void* d_out, int out_size, void* d_ws, size_t ws_size,
                              hipStream_t stream) {
  (void)in_sizes; (void)n_in; (void)d_ws; (void)ws_size; (void)out_size;
  const float* x = (const float*)d_in[0];
  // tuple output, concatenated flat: [B,C,K] f32 values, then [B,C,K,2] i32 coords
  float* out_val  = (float*)d_out;
  int*   out_coor = (int*)d_out + ROWS * TOPK;
  topk_l2_rows_kernel<<<ROWS, NT, 0, stream>>>(x, out_val, out_coor);
}